// EfficientDet_55989193671252
// MI455X (gfx1250) — compile-verified
//
#include <hip/hip_runtime.h>
#include <stdint.h>

// ---------------------------------------------------------------------------
// EfficientDet post-process for MI455X (gfx1250).
// CDNA5-specific paths:
//   * k_hist    : Tensor Data Mover (tensor_load_to_lds + s_wait_tensorcnt)
//                 double-buffered 16KB tile streaming into LDS.
//   * k_compact : async global->LDS data mover (ASYNCcnt) double buffering.
//   * k_nms     : wave32 shuffle-reduction argmax.
// ---------------------------------------------------------------------------

#define NUM_CLASSES 90
#define NUM_ANCH    9
#define MAXDET      5000
#define MAXOUT      100
#define BATCH       16
#define NBUCKET     4096
#define CAP         16384
#define SIGMA       0.5f
#define TILE_DW     4096            // dwords per TDM tile (16 KB)

typedef unsigned int v4u __attribute__((ext_vector_type(4)));
typedef int          v8i __attribute__((ext_vector_type(8)));
typedef int          v4i __attribute__((ext_vector_type(4)));

struct Args {
  const float* cls[5];
  const float* box[5];
  const float* anchors;
  int n4[5];      // per-level float4 count per image (810*HW/4)
  int hw[5];      // H*W per level (power of two)
  int loghw[5];   // log2(hw)
  int aoff[6];    // anchor offsets per level, aoff[5] = total anchors
  uint32_t* ghist;     // [B][NBUCKET]
  int*      cnt;       // [B][2]  (hiCnt, ovfCnt)
  int*      thrInfo;   // [B][4]  (thr, countAbove, need)
  float*    topScore;  // [B][MAXDET]  logit -> later sigmoid score
  int*      topAnchor; // [B][MAXDET]
  int*      topClass;  // [B][MAXDET]
  float*    boxes;     // [B][MAXDET][4]  xyxy
  uint32_t* ovfKey;    // [B][CAP]
  uint32_t* ovfPay;    // [B][CAP]
  float*    out;       // [B][MAXOUT][6]
};

// monotone float -> uint key (order-preserving), and inverse
__device__ __forceinline__ uint32_t ordbits(float f) {
  uint32_t u = __float_as_uint(f);
  uint32_t m = (uint32_t)(((int32_t)u) >> 31) | 0x80000000u;
  return u ^ m;
}
__device__ __forceinline__ float inv_ordbits(uint32_t k) {
  uint32_t m = (uint32_t)(((int32_t)(~k)) >> 31) | 0x80000000u;
  return __uint_as_float(k ^ m);
}

// Issue one TDM load of `ndwords` 4-byte elements (single row) into LDS.
// D# per CDNA5 ISA:
//   group0: [1:0]=count=1, [63:32]=lds_addr, [120:64]=global_addr, [127:126]=type 2
//   group1: [17:16]=data_size(2 => 4B), tensor_dim0=ndwords, tensor_dim1=1,
//           tile_dim0=ndwords (tile == tensor row, so no OOB zero-fill)
// 6-arg builtin form (clang-23 / therock): (v4u, v8i, v4i, v4i, v8i, i32 cpol)
__device__ __forceinline__ void tdm_load(void* lds_dst, const void* gsrc, int ndwords) {
  unsigned long long ga = (unsigned long long)(uintptr_t)gsrc;
  uint32_t lds = (uint32_t)(uintptr_t)lds_dst;      // flat addr low 32b == LDS offset
  uint32_t td0 = (uint32_t)ndwords;                  // tensor_dim0
  v4u g0;
  g0[0] = 1u;                                        // count=1 user descriptor
  g0[1] = lds;                                       // lds_addr
  g0[2] = (uint32_t)ga;                              // global_addr[31:0]
  g0[3] = (uint32_t)(ga >> 32) | (2u << 30);         // global_addr[56:32] | type=2
  v8i g1;
  g1[0] = (int)(2u << 16);                           // wg_mask=0, data_size=4B
  g1[1] = (int)((td0 & 0xFFFFu) << 16);              // tensor_dim0[15:0]
  g1[2] = (int)(((td0 >> 16) & 0xFFFFu) | (1u << 16)); // tensor_dim0[31:16] | tensor_dim1=1
  g1[3] = (int)(td0 << 16);                          // tensor_dim1[31:16]=0 | tile_dim0
  g1[4] = 0;                                         // tile_dim1=0, tile_dim2=0 (unused)
  g1[5] = 0; g1[6] = 0; g1[7] = 0;                   // strides unused (one row)
  v4i z4 = {0, 0, 0, 0};
  v8i z8 = {0, 0, 0, 0, 0, 0, 0, 0};
  __builtin_amdgcn_tensor_load_to_lds(g0, g1, z4, z4, z8, 0);
}

// -------------------- pass 1: 12-bit key histogram (TDM streaming) --------
__global__ __launch_bounds__(256) void k_hist(Args A) {
  __shared__ uint32_t hist[NBUCKET];        // 16 KB
  __shared__ float    stage[2][TILE_DW];    // 32 KB double buffer
  const int tid = threadIdx.x;
  const int wv  = tid >> 5;
  const int img = blockIdx.y, lvl = blockIdx.z;
  const int ndw = A.n4[lvl] * 4;
  const float* src = A.cls[lvl] + (size_t)img * ndw;

  for (int i = tid; i < NBUCKET; i += 256) hist[i] = 0u;
  __syncthreads();

  const int nt   = (ndw + TILE_DW - 1) / TILE_DW;
  const int t0   = blockIdx.x;
  const int step = gridDim.x;

  if (t0 < nt && wv == 0) {
    int c0 = min(TILE_DW, ndw - t0 * TILE_DW);
    tdm_load(&stage[0][0], src + (size_t)t0 * TILE_DW, c0);
  }
  int buf = 0;
  for (int t = t0; t < nt; t += step) {
    const int cnt = min(TILE_DW, ndw - t * TILE_DW);   // always multiple of 4
    const int tn = t + step;
    const bool nhave = tn < nt;
    if (wv == 0) {
      if (nhave) {
        int cn = min(TILE_DW, ndw - tn * TILE_DW);
        tdm_load(&stage[buf ^ 1][0], src + (size_t)tn * TILE_DW, cn);
        __builtin_amdgcn_s_wait_tensorcnt((short)1);   // tile t done, t+1 in flight
      } else {
        __builtin_amdgcn_s_wait_tensorcnt((short)0);
      }
    }
    __syncthreads();                                   // publish tile to block
    const float4* sp = (const float4*)&stage[buf][0];
    for (int i4 = tid; i4 * 4 < cnt; i4 += 256) {
      float4 v = sp[i4];
      atomicAdd(&hist[ordbits(v.x) >> 20], 1u);
      atomicAdd(&hist[ordbits(v.y) >> 20], 1u);
      atomicAdd(&hist[ordbits(v.z) >> 20], 1u);
      atomicAdd(&hist[ordbits(v.w) >> 20], 1u);
    }
    __syncthreads();                                   // buffer reusable
    buf ^= 1;
  }
  uint32_t* gh = A.ghist + (size_t)img * NBUCKET;
  for (int i = tid; i < NBUCKET; i += 256)
    if (hist[i]) atomicAdd(&gh[i], hist[i]);
}

// -------------------- pass 2: threshold bucket per image ------------------
__global__ __launch_bounds__(256) void k_scan(Args A) {
  const int img = blockIdx.x;
  __shared__ uint32_t h[NBUCKET];
  for (int i = threadIdx.x; i < NBUCKET; i += 256)
    h[i] = A.ghist[(size_t)img * NBUCKET + i];
  __syncthreads();
  if (threadIdx.x == 0) {
    uint32_t cum = 0; int thr = 0, cA = 0, need = MAXDET;
    for (int bkt = NBUCKET - 1; bkt >= 0; --bkt) {
      uint32_t c = h[bkt];
      if (cum + c >= (uint32_t)MAXDET) { thr = bkt; cA = (int)cum; need = MAXDET - (int)cum; break; }
      cum += c;
    }
    A.thrInfo[img * 4 + 0] = thr;
    A.thrInfo[img * 4 + 1] = cA;
    A.thrInfo[img * 4 + 2] = need;
  }
}

// -------------------- pass 3: compaction (async-to-LDS streaming) ---------
// Each wave streams its own tiles through a private double-buffered LDS stage
// using global_load_async_to_lds_b128 (ASYNCcnt-tracked).
__global__ __launch_bounds__(256) void k_compact(Args A) {
  __shared__ float4 stage[2][8][32];    // per-wave private slices, 8 KB
  const int tid  = threadIdx.x;
  const int lane = tid & 31;
  const int wv   = tid >> 5;
  const int img  = blockIdx.y, lvl = blockIdx.z;
  const int n4   = A.n4[lvl], loghw = A.loghw[lvl];
  const int hwm  = A.hw[lvl] - 1, aoff = A.aoff[lvl];
  const float4* src = (const float4*)A.cls[lvl] + (size_t)img * n4;
  const int thr = A.thrInfo[img * 4 + 0];
  int* hiCnt = &A.cnt[img * 2 + 0];
  int* ovCnt = &A.cnt[img * 2 + 1];
  float* tS = A.topScore  + (size_t)img * MAXDET;
  int*   tA = A.topAnchor + (size_t)img * MAXDET;
  int*   tC = A.topClass  + (size_t)img * MAXDET;
  uint32_t* oK = A.ovfKey + (size_t)img * CAP;
  uint32_t* oP = A.ovfPay + (size_t)img * CAP;

  const uint32_t lds0 = (uint32_t)(uintptr_t)&stage[0][wv][lane];
  const uint32_t lds1 = (uint32_t)(uintptr_t)&stage[1][wv][lane];

  const int W = gridDim.x * 8;
  const int g = blockIdx.x * 8 + wv;
  int  base = g * 32;
  bool have = base < n4;
  if (have && (base + lane) < n4) {
    unsigned long long ga = (unsigned long long)(uintptr_t)(src + base + lane);
    asm volatile("global_load_async_to_lds_b128 %0, %1, off"
                 :: "v"(lds0), "v"(ga) : "memory");
  }
  int buf = 0;
  while (have) {
    const int nbase = base + 32 * W;
    const bool nhave = nbase < n4;
    if (nhave && (nbase + lane) < n4) {
      uint32_t dst = buf ? lds0 : lds1;
      unsigned long long ga = (unsigned long long)(uintptr_t)(src + nbase + lane);
      asm volatile("global_load_async_to_lds_b128 %0, %1, off"
                   :: "v"(dst), "v"(ga) : "memory");
    }
    if (nhave) { asm volatile("s_wait_asynccnt 1" ::: "memory"); }
    else       { asm volatile("s_wait_asynccnt 0" ::: "memory"); }
    const int i4 = base + lane;
    if (i4 < n4) {
      float4 v = stage[buf][wv][lane];
      const int e0 = i4 << 2;
      const int ch = e0 >> loghw;       // channel = a*90 + c
      const int s0 = e0 & hwm;          // spatial index of element 0
      const int a  = ch / NUM_CLASSES;
      const int c  = ch - a * NUM_CLASSES;
      float f[4] = {v.x, v.y, v.z, v.w};
#pragma unroll
      for (int j = 0; j < 4; ++j) {
        const uint32_t k = ordbits(f[j]);
        const int bkt = (int)(k >> 20);
        if (bkt < thr) continue;
        const int anchor = aoff + (s0 + j) * NUM_ANCH + a;
        if (bkt > thr) {
          int slot = atomicAdd(hiCnt, 1);
          if (slot < MAXDET) { tS[slot] = f[j]; tA[slot] = anchor; tC[slot] = c; }
        } else {
          int o = atomicAdd(ovCnt, 1);
          if (o < CAP) { oK[o] = k; oP[o] = (uint32_t)anchor * NUM_CLASSES + c; }
        }
      }
    }
    base = nbase; buf ^= 1; have = nhave;
  }
}

// -------------------- pass 4: boundary refine (next 12 key bits) ----------
__global__ __launch_bounds__(256) void k_refine(Args A) {
  const int img = blockIdx.x;
  __shared__ uint32_t h2[NBUCKET];
  __shared__ int thr2s, cA2s, need2s, c1, c2;
  const int tid  = threadIdx.x;
  const int base = A.thrInfo[img * 4 + 1];
  const int need = A.thrInfo[img * 4 + 2];
  int n = A.cnt[img * 2 + 1]; if (n > CAP) n = CAP;
  float* tS = A.topScore  + (size_t)img * MAXDET;
  int*   tA = A.topAnchor + (size_t)img * MAXDET;
  int*   tC = A.topClass  + (size_t)img * MAXDET;
  const uint32_t* oK = A.ovfKey + (size_t)img * CAP;
  const uint32_t* oP = A.ovfPay + (size_t)img * CAP;

  for (int k = base + tid; k < MAXDET; k += 256) { tS[k] = -1e30f; tA[k] = 0; tC[k] = 0; }
  for (int i = tid; i < NBUCKET; i += 256) h2[i] = 0u;
  if (tid == 0) { c1 = 0; c2 = 0; }
  __syncthreads();
  for (int i = tid; i < n; i += 256) atomicAdd(&h2[(oK[i] >> 8) & 0xFFFu], 1u);
  __syncthreads();
  if (tid == 0) {
    uint32_t cum = 0; int thr2 = -2, cA2 = 0, need2 = 0;   // -2 => take all
    for (int bkt = NBUCKET - 1; bkt >= 0; --bkt) {
      uint32_t c = h2[bkt];
      if ((int)(cum + c) >= need) { thr2 = bkt; cA2 = (int)cum; need2 = need - (int)cum; break; }
      cum += c;
    }
    thr2s = thr2; cA2s = cA2; need2s = need2;
  }
  __syncthreads();
  const int thr2 = thr2s, cA2 = cA2s, need2 = need2s;
  for (int i = tid; i < n; i += 256) {
    const uint32_t k = oK[i];
    const int b2 = (int)((k >> 8) & 0xFFFu);
    int slot = -1;
    if (b2 > thr2)       slot = base + atomicAdd(&c1, 1);
    else if (b2 == thr2) { int o = atomicAdd(&c2, 1); if (o < need2) slot = base + cA2 + o; }
    if (slot >= 0 && slot < MAXDET) {
      const uint32_t p = oP[i];
      tS[slot] = inv_ordbits(k);
      tA[slot] = (int)(p / NUM_CLASSES);
      tC[slot] = (int)(p % NUM_CLASSES);
    }
  }
}

// -------------------- pass 5: gather + decode + sigmoid -------------------
__global__ __launch_bounds__(256) void k_decode(Args A) {
  const int img = blockIdx.y;
  const int k = blockIdx.x * 256 + threadIdx.x;
  if (k >= MAXDET) return;
  const size_t idx = (size_t)img * MAXDET + k;
  const int anchor = A.topAnchor[idx];
  int lvl = 0;
  while (lvl < 4 && anchor >= A.aoff[lvl + 1]) ++lvl;
  const int local = anchor - A.aoff[lvl];
  const int s = local / NUM_ANCH, a = local - s * NUM_ANCH;
  const int HW = A.hw[lvl];
  const float* bp = A.box[lvl] + ((size_t)img * NUM_ANCH * 4 + (size_t)a * 4) * HW + s;
  const float ty = bp[0], tx = bp[HW], th = bp[2 * HW], tw = bp[3 * HW];
  const float* an = A.anchors + (size_t)anchor * 4;        // y1,x1,y2,x2
  const float y1 = an[0], x1 = an[1], y2 = an[2], x2 = an[3];
  const float ya = (y1 + y2) * 0.5f, xa = (x1 + x2) * 0.5f;
  const float ha = y2 - y1,          wa = x2 - x1;
  const float w  = __expf(tw) * wa,  h  = __expf(th) * ha;
  const float yc = ty * ha + ya,     xc = tx * wa + xa;
  float* bx = A.boxes + idx * 4;
  bx[0] = xc - w * 0.5f; bx[1] = yc - h * 0.5f;
  bx[2] = xc + w * 0.5f; bx[3] = yc + h * 0.5f;
  const float logit = A.topScore[idx];
  A.topScore[idx] = 1.0f / (1.0f + __expf(-logit));
}

// -------------------- pass 6: gaussian soft-NMS ---------------------------
__global__ __launch_bounds__(1024) void k_nms(Args A) {
  __shared__ float sc[MAXDET];       // 20 KB working scores
  __shared__ float pV[32];
  __shared__ int   pI[32];
  __shared__ float selB[4];
  __shared__ int   selI;
  const int tid = threadIdx.x, lane = tid & 31, wv = tid >> 5;
  const int img = blockIdx.x;
  const float* gb = A.boxes + (size_t)img * MAXDET * 4;

  for (int i = tid; i < MAXDET; i += 1024) sc[i] = A.topScore[(size_t)img * MAXDET + i];
  __syncthreads();

  for (int it = 0; it < MAXOUT; ++it) {
    // block argmax (first-index tie-break), wave32 shuffle reduction
    float v = -3.0e38f; int idx = 0;
    for (int i = tid; i < MAXDET; i += 1024) { float s = sc[i]; if (s > v) { v = s; idx = i; } }
#pragma unroll
    for (int o = 16; o > 0; o >>= 1) {
      float ov = __shfl_xor(v, o, 32); int oi = __shfl_xor(idx, o, 32);
      if (ov > v || (ov == v && oi < idx)) { v = ov; idx = oi; }
    }
    if (lane == 0) { pV[wv] = v; pI[wv] = idx; }
    __syncthreads();
    if (wv == 0) {
      v = pV[lane]; idx = pI[lane];
#pragma unroll
      for (int o = 16; o > 0; o >>= 1) {
        float ov = __shfl_xor(v, o, 32); int oi = __shfl_xor(idx, o, 32);
        if (ov > v || (ov == v && oi < idx)) { v = ov; idx = oi; }
      }
      if (lane == 0) {
        selI = idx;
        const float* b4 = gb + (size_t)idx * 4;
        float b0 = b4[0], b1 = b4[1], b2 = b4[2], b3 = b4[3];
        selB[0] = b0; selB[1] = b1; selB[2] = b2; selB[3] = b3;
        float* o6 = A.out + ((size_t)img * MAXOUT + it) * 6;
        o6[0] = b0; o6[1] = b1; o6[2] = b2; o6[3] = b3;
        o6[4] = v;
        o6[5] = (float)A.topClass[(size_t)img * MAXDET + idx];
      }
    }
    __syncthreads();
    const float p0 = selB[0], p1 = selB[1], p2 = selB[2], p3 = selB[3];
    const int sel = selI;
    const float area1 = (p2 - p0) * (p3 - p1);
    for (int i = tid; i < MAXDET; i += 1024) {
      const float* c4 = gb + (size_t)i * 4;
      const float c0 = c4[0], c1v = c4[1], c2v = c4[2], c3 = c4[3];
      const float xx1 = fmaxf(p0, c0),  yy1 = fmaxf(p1, c1v);
      const float xx2 = fminf(p2, c2v), yy2 = fminf(p3, c3);
      const float inter = fmaxf(xx2 - xx1, 0.0f) * fmaxf(yy2 - yy1, 0.0f);
      const float a2 = (c2v - c0) * (c3 - c1v);
      const float iou = inter / (area1 + a2 - inter + 1e-8f);
      const float decay = __expf(-(iou * iou) * (1.0f / SIGMA));
      sc[i] = (i == sel) ? -3.0e38f : sc[i] * decay;
    }
    __syncthreads();
  }
}

// ---------------------------------------------------------------------------
extern "C" void kernel_launch(void* const* d_in, const int* in_sizes, int n_in,
                              void* d_out, int out_size, void* d_ws, size_t ws_size,
                              hipStream_t stream) {
  (void)in_sizes; (void)n_in; (void)out_size; (void)ws_size;
  Args A;
  for (int l = 0; l < 5; ++l) {
    A.cls[l] = (const float*)d_in[2 * l];        // dict order: cls_l{i}, box_l{i} interleaved
    A.box[l] = (const float*)d_in[2 * l + 1];
  }
  A.anchors = (const float*)d_in[10];
  static const int HW[5]  = {4096, 1024, 256, 64, 16};
  static const int LOG[5] = {12, 10, 8, 6, 4};
  int aoff = 0;
  for (int l = 0; l < 5; ++l) {
    A.hw[l] = HW[l]; A.loghw[l] = LOG[l];
    A.n4[l] = (810 * HW[l]) / 4;
    A.aoff[l] = aoff; aoff += HW[l] * NUM_ANCH;
  }
  A.aoff[5] = aoff;  // 49104

  char* ws = (char*)d_ws;
  size_t off = 0;
  auto alloc = [&](size_t bytes) -> void* {
    void* p = ws + off;
    off = (off + bytes + 255) & ~(size_t)255;
    return p;
  };
  A.ghist     = (uint32_t*)alloc((size_t)BATCH * NBUCKET * 4);
  A.cnt       = (int*)     alloc((size_t)BATCH * 2 * 4);
  A.thrInfo   = (int*)     alloc((size_t)BATCH * 4 * 4);
  A.topScore  = (float*)   alloc((size_t)BATCH * MAXDET * 4);
  A.topAnchor = (int*)     alloc((size_t)BATCH * MAXDET * 4);
  A.topClass  = (int*)     alloc((size_t)BATCH * MAXDET * 4);
  A.boxes     = (float*)   alloc((size_t)BATCH * MAXDET * 4 * 4);
  A.ovfKey    = (uint32_t*)alloc((size_t)BATCH * CAP * 4);
  A.ovfPay    = (uint32_t*)alloc((size_t)BATCH * CAP * 4);
  A.out       = (float*)d_out;

  (void)hipMemsetAsync(A.ghist, 0, (size_t)BATCH * NBUCKET * 4, stream);
  (void)hipMemsetAsync(A.cnt,   0, (size_t)BATCH * 2 * 4, stream);

  k_hist   <<<dim3(32, BATCH, 5), dim3(256), 0, stream>>>(A);
  k_scan   <<<dim3(BATCH),        dim3(256), 0, stream>>>(A);
  k_compact<<<dim3(64, BATCH, 5), dim3(256), 0, stream>>>(A);
  k_refine <<<dim3(BATCH),        dim3(256), 0, stream>>>(A);
  k_decode <<<dim3((MAXDET + 255) / 256, BATCH), dim3(256), 0, stream>>>(A);
  k_nms    <<<dim3(BATCH),        dim3(1024), 0, stream>>>(A);
}